// GraphSAGE_9706626089388
// MI455X (gfx1250) — compile-verified
//
#include <hip/hip_runtime.h>
#include <math.h>

typedef __attribute__((ext_vector_type(2))) float v2f;
typedef __attribute__((ext_vector_type(8))) float v8f;

// ---------------------------------------------------------------------------
// Zero a range of floats (deg + agg1 + agg2 accumulators).
// ---------------------------------------------------------------------------
__global__ void k_zero(float* __restrict__ p, size_t n) {
  size_t i = (size_t)blockIdx.x * blockDim.x + threadIdx.x;
  size_t stride = (size_t)gridDim.x * blockDim.x;
  for (; i < n; i += stride) p[i] = 0.0f;
}

// ---------------------------------------------------------------------------
// Weight transpose: Wt[n*K + k] = W[k*N + n]   (row-major KxN -> NxK)
// Makes B-fragment loads contiguous b64 with immediate offsets in the GEMM.
// ---------------------------------------------------------------------------
__global__ void k_transpose(const float* __restrict__ W, float* __restrict__ Wt,
                            int K, int N) {
  int idx = blockIdx.x * blockDim.x + threadIdx.x;
  if (idx >= K * N) return;
  int n = idx / K;
  int k = idx - n * K;
  Wt[idx] = W[(size_t)k * N + n];
}

// ---------------------------------------------------------------------------
// Edge scatter: one wave32 per edge. Lane l handles features [4l, 4l+4):
// float4 gather from feat[src], 4 relaxed agent-scope f32 atomic adds into
// agg[dst] (global_atomic_add_f32; accumulator stays L2-resident).
// ---------------------------------------------------------------------------
__global__ __launch_bounds__(256)
void k_scatter(const float* __restrict__ feat, const int* __restrict__ src,
               const int* __restrict__ dst, float* __restrict__ agg,
               float* __restrict__ deg, int nE) {
  int wave = (int)((blockIdx.x * blockDim.x + threadIdx.x) >> 5);
  int lane = threadIdx.x & 31;
  if (wave >= nE) return;
  int s = src[wave];
  int d = dst[wave];
  if (deg != nullptr && lane == 0) {
    __hip_atomic_fetch_add(&deg[d], 1.0f, __ATOMIC_RELAXED, __HIP_MEMORY_SCOPE_AGENT);
  }
  const float4 v = *reinterpret_cast<const float4*>(feat + (size_t)s * 128 + lane * 4);
  float* base = agg + (size_t)d * 128 + lane * 4;
  __hip_atomic_fetch_add(base + 0, v.x, __ATOMIC_RELAXED, __HIP_MEMORY_SCOPE_AGENT);
  __hip_atomic_fetch_add(base + 1, v.y, __ATOMIC_RELAXED, __HIP_MEMORY_SCOPE_AGENT);
  __hip_atomic_fetch_add(base + 2, v.z, __ATOMIC_RELAXED, __HIP_MEMORY_SCOPE_AGENT);
  __hip_atomic_fetch_add(base + 3, v.w, __ATOMIC_RELAXED, __HIP_MEMORY_SCOPE_AGENT);
}

// ---------------------------------------------------------------------------
// Fused SAGE layer GEMM via V_WMMA_F32_16X16X4_F32 (wave32), K = 128 fixed:
//   out = sigmoid( (agg/deg) @ Wl + self @ Wr + bias )
// Weights are pre-transposed (Wt[n][k], stride K=128), so each B fragment is
// one contiguous global_load_b64 at an immediate offset from a per-lane base.
// K is processed in chunks of 32 with 8 B-fragments preloaded per chunk so
// up to 8 loads stay in flight ahead of the WMMAs.
// A frag (16x4 f32):  lane m+16h -> {A[m][2h], A[m][2h+1]}   (ds_load_b64)
// B frag (4x16 f32):  lane n+16h -> {Wt[n][2h], Wt[n][2h+1]} (global b64)
// D (16x16 f32, v8f): row = m0 + v + 8*(lane>>4), col = j0 + (lane&15)
// ---------------------------------------------------------------------------
__global__ __launch_bounds__(256)
void k_sage_gemm(const float* __restrict__ agg, const float* __restrict__ deg,
                 const float* __restrict__ selff,
                 const float* __restrict__ WtL, const float* __restrict__ WtR,
                 const float* __restrict__ bias, float* __restrict__ out,
                 int Ncols) {
  __shared__ float lAm[16][132];   // mean-aggregated tile (pad 132 -> no conflicts)
  __shared__ float lAs[16][132];   // self-feature tile
  __shared__ float rdeg[16];

  const int K = 128;
  const int m0 = blockIdx.x * 16;

  if (threadIdx.x < 16) {
    rdeg[threadIdx.x] = 1.0f / fmaxf(deg[m0 + threadIdx.x], 1.0f);
  }
  __syncthreads();

  for (int i = threadIdx.x; i < 16 * K; i += 256) {
    int r = i >> 7;
    int c = i & 127;
    size_t off = (size_t)(m0 + r) * K + c;
    lAm[r][c] = agg[off] * rdeg[r];
    lAs[r][c] = selff[off];
  }
  __syncthreads();

  const int wid  = threadIdx.x >> 5;
  const int lane = threadIdx.x & 31;
  const int mn   = lane & 15;   // "m" for A frags, "n" for B/D frags
  const int hh   = lane >> 4;   // K-pair / row-group selector
  const int ntiles = Ncols >> 4;

  for (int jt = wid; jt < ntiles; jt += 8) {
    const int j0 = jt << 4;
    v8f acc;
    const float bv = bias[j0 + mn];
#pragma unroll
    for (int v = 0; v < 8; ++v) acc[v] = bv;

    // per-lane B base pointers into transposed weights (stride K = 128)
    const float* bl = WtL + (size_t)(j0 + mn) * K + 2 * hh;
    const float* br = WtR + (size_t)(j0 + mn) * K + 2 * hh;

    // (agg/deg) @ Wl : K chunks of 32, 8 B-frags in flight
    for (int kc = 0; kc < K; kc += 32) {
      v2f bfrag[8];
#pragma unroll
      for (int u = 0; u < 8; ++u) {
        bfrag[u] = *reinterpret_cast<const v2f*>(bl + kc + 4 * u);
      }
#pragma unroll
      for (int u = 0; u < 8; ++u) {
        const int kb = kc + 4 * u;
        v2f a = *reinterpret_cast<const v2f*>(&lAm[mn][kb + 2 * hh]);
        acc = __builtin_amdgcn_wmma_f32_16x16x4_f32(false, a, false, bfrag[u],
                                                    (short)0, acc, false, false);
      }
    }
    // self @ Wr
    for (int kc = 0; kc < K; kc += 32) {
      v2f bfrag[8];
#pragma unroll
      for (int u = 0; u < 8; ++u) {
        bfrag[u] = *reinterpret_cast<const v2f*>(br + kc + 4 * u);
      }
#pragma unroll
      for (int u = 0; u < 8; ++u) {
        const int kb = kc + 4 * u;
        v2f a = *reinterpret_cast<const v2f*>(&lAs[mn][kb + 2 * hh]);
        acc = __builtin_amdgcn_wmma_f32_16x16x4_f32(false, a, false, bfrag[u],
                                                    (short)0, acc, false, false);
      }
    }

    // fused sigmoid + store
#pragma unroll
    for (int v = 0; v < 8; ++v) {
      float z = acc[v];
      float sg = 1.0f / (1.0f + __expf(-z));
      int row = m0 + v + 8 * hh;
      out[(size_t)row * Ncols + j0 + mn] = sg;
    }
  }
}

// ---------------------------------------------------------------------------
// In-place row L2 normalization: blockDim.x == 128 (one block per node row).
// ---------------------------------------------------------------------------
__global__ __launch_bounds__(128)
void k_l2norm(float* __restrict__ h) {
  __shared__ float red[128];
  const int D = 128;
  const int t = threadIdx.x;
  size_t base = (size_t)blockIdx.x * D;
  float v = h[base + t];
  red[t] = v * v;
  __syncthreads();
  for (int s = D >> 1; s > 0; s >>= 1) {
    if (t < s) red[t] += red[t + s];
    __syncthreads();
  }
  float nrm = fmaxf(sqrtf(red[0]), 1e-12f);
  h[base + t] = v / nrm;
}

// ---------------------------------------------------------------------------
// Fused row L2 normalize + log_softmax for the output layer (D = 256).
// ---------------------------------------------------------------------------
__global__ __launch_bounds__(256)
void k_post2(float* __restrict__ out) {
  __shared__ float red[256];
  const int D = 256;
  const int t = threadIdx.x;
  size_t base = (size_t)blockIdx.x * D;
  float v = out[base + t];

  red[t] = v * v;
  __syncthreads();
  for (int s = D >> 1; s > 0; s >>= 1) {
    if (t < s) red[t] += red[t + s];
    __syncthreads();
  }
  float nrm = fmaxf(sqrtf(red[0]), 1e-12f);
  __syncthreads();
  float h = v / nrm;

  red[t] = h;
  __syncthreads();
  for (int s = D >> 1; s > 0; s >>= 1) {
    if (t < s) red[t] = fmaxf(red[t], red[t + s]);
    __syncthreads();
  }
  float mx = red[0];
  __syncthreads();

  float e = __expf(h - mx);
  red[t] = e;
  __syncthreads();
  for (int s = D >> 1; s > 0; s >>= 1) {
    if (t < s) red[t] += red[t + s];
    __syncthreads();
  }
  float lse = logf(red[0]);
  out[base + t] = h - mx - lse;
}

// ---------------------------------------------------------------------------
// kernel_launch
// Inputs: x, edge_index, W1_l, W1_r, b1, W2_l, W2_r, b2  (all f32 / i32)
// ---------------------------------------------------------------------------
extern "C" void kernel_launch(void* const* d_in, const int* in_sizes, int n_in,
                              void* d_out, int out_size, void* d_ws, size_t ws_size,
                              hipStream_t stream) {
  const float* x   = (const float*)d_in[0];
  const int*   ei  = (const int*)d_in[1];
  const float* W1l = (const float*)d_in[2];
  const float* W1r = (const float*)d_in[3];
  const float* b1  = (const float*)d_in[4];
  const float* W2l = (const float*)d_in[5];
  const float* W2r = (const float*)d_in[6];
  const float* b2  = (const float*)d_in[7];
  float* out = (float*)d_out;

  const int D_IN = 128;
  const int N = in_sizes[0] / D_IN;     // 50000
  const int E = in_sizes[1] / 2;        // 800000
  const int D_HID = 128;
  const int D_OUT = 256;

  const int* src = ei;
  const int* dst = ei + E;

  // workspace layout (f32)
  float* ws = (float*)d_ws;
  const size_t Npad = ((size_t)N + 255) & ~(size_t)255;
  const size_t ND = (size_t)N * D_IN;
  float* deg   = ws;               // N
  float* agg1  = ws + Npad;        // N*128
  float* agg2  = agg1 + ND;        // N*128
  float* h1    = agg2 + ND;        // N*128
  float* Wt1l  = h1 + ND;          // 128*128
  float* Wt1r  = Wt1l + 128 * 128; // 128*128
  float* Wt2l  = Wt1r + 128 * 128; // 128*256
  float* Wt2r  = Wt2l + 128 * 256; // 128*256

  // 1) zero degree + both scatter accumulators
  size_t nzero = Npad + 2 * ND;
  k_zero<<<4096, 256, 0, stream>>>(ws, nzero);

  // 1b) transpose weights (tiny; enables b64 immediate-offset B loads)
  k_transpose<<<(128 * 128 + 255) / 256, 256, 0, stream>>>(W1l, Wt1l, 128, D_HID);
  k_transpose<<<(128 * 128 + 255) / 256, 256, 0, stream>>>(W1r, Wt1r, 128, D_HID);
  k_transpose<<<(128 * 256 + 255) / 256, 256, 0, stream>>>(W2l, Wt2l, 128, D_OUT);
  k_transpose<<<(128 * 256 + 255) / 256, 256, 0, stream>>>(W2r, Wt2r, 128, D_OUT);

  // 2) layer-1 scatter-sum of x onto agg1 (+ degree)
  {
    int waves_per_block = 256 / 32;
    int blocks = (E + waves_per_block - 1) / waves_per_block;
    k_scatter<<<blocks, 256, 0, stream>>>(x, src, dst, agg1, deg, E);
  }

  // 3) layer 1: h1 = sigmoid(mean@W1l + x@W1r + b1)   [WMMA f32]
  k_sage_gemm<<<N / 16, 256, 0, stream>>>(agg1, deg, x, Wt1l, Wt1r, b1, h1, D_HID);

  // 4) row L2 normalize h1 in place
  k_l2norm<<<N, 128, 0, stream>>>(h1);

  // 5) layer-2 scatter-sum of h1 onto agg2 (degree already known)
  {
    int waves_per_block = 256 / 32;
    int blocks = (E + waves_per_block - 1) / waves_per_block;
    k_scatter<<<blocks, 256, 0, stream>>>(h1, src, dst, agg2, nullptr, E);
  }

  // 6) layer 2: out = sigmoid(mean2@W2l + h1@W2r + b2)   [WMMA f32]
  k_sage_gemm<<<N / 16, 256, 0, stream>>>(agg2, deg, h1, Wt2l, Wt2r, b2, out, D_OUT);

  // 7) row L2 normalize + log_softmax in place on out
  k_post2<<<N, 256, 0, stream>>>(out);
}